// LightGCN_11682311045932
// MI455X (gfx1250) — compile-verified
//
#include <hip/hip_runtime.h>
#include <hip/hip_bf16.h>

// ---------------------------------------------------------------------------
// LightGCN forward (3-hop SpMM propagation + InfoNCE loss) for gfx1250.
// SpMM phase: L2-resident (38.4MB << 192MB L2), edge-per-wave, f32 atomics.
// Scoring phase: V_WMMA_F32_16X16X4_F32 chains computing diag(U_blk @ V_blk^T).
// ---------------------------------------------------------------------------

typedef float v2f __attribute__((ext_vector_type(2)));
typedef float v8f __attribute__((ext_vector_type(8)));

#define DIM        64
#define HOPS       3
#define NUM_NEG    8

// ---------------------------------------------------------------------------
// Kernel 1: concat user/item embeddings -> x0, and acc = x0.  float4-vectorized.
// ---------------------------------------------------------------------------
__global__ void lgcn_init_concat(const float4* __restrict__ user_emb,
                                 const float4* __restrict__ item_emb,
                                 float4* __restrict__ x,
                                 float4* __restrict__ acc,
                                 int user_vec4, int total_vec4) {
    int i = blockIdx.x * blockDim.x + threadIdx.x;
    if (i >= total_vec4) return;
    float4 v = (i < user_vec4) ? user_emb[i] : item_emb[i - user_vec4];
    x[i]   = v;
    acc[i] = v;
}

// ---------------------------------------------------------------------------
// Kernel 2: SpMM  y += A * x  (COO, one edge per wave32).
// Edge index is wave-uniform -> compiler scalarizes the metadata loads.
// Each lane handles float2 (32 lanes * 8B = 256B per row, fully coalesced).
// Scatter via global_atomic_add_f32 (all traffic L2-resident).
// ---------------------------------------------------------------------------
__global__ void lgcn_spmm(const float* __restrict__ x,
                          float* __restrict__ y,
                          const float* __restrict__ vals,
                          const int* __restrict__ rows,
                          const int* __restrict__ cols,
                          int nnz) {
    int wave = blockIdx.x * (blockDim.x >> 5) + (threadIdx.x >> 5); // uniform in wave
    int lane = threadIdx.x & 31;
    if (wave >= nnz) return;
    int   col = cols[wave];
    int   row = rows[wave];
    float v   = vals[wave];
    const v2f* xs = (const v2f*)(x + (size_t)col * DIM);
    v2f m = xs[lane];
    float* yr = y + (size_t)row * DIM + lane * 2;
    atomicAdd(yr + 0, m.x * v);
    atomicAdd(yr + 1, m.y * v);
}

// ---------------------------------------------------------------------------
// Kernel 3: acc += x   (float4-vectorized elementwise)
// ---------------------------------------------------------------------------
__global__ void lgcn_accum(float4* __restrict__ acc,
                           const float4* __restrict__ x,
                           int total_vec4) {
    int i = blockIdx.x * blockDim.x + threadIdx.x;
    if (i >= total_vec4) return;
    float4 a = acc[i];
    float4 b = x[i];
    a.x += b.x; a.y += b.y; a.z += b.z; a.w += b.w;
    acc[i] = a;
}

// ---------------------------------------------------------------------------
// Kernel 4: InfoNCE scoring via WMMA.
// One wave32 per group of 16 batch entries. For each of 9 partner sets
// (1 positive item + 8 negatives) compute diag(U[16x64] @ V[16x64]^T) with a
// chain of 16 V_WMMA_F32_16X16X4_F32, scale by 1/16 (folds light_out /4 on
// both operands), accumulate exp-sums, atomicAdd mean loss into d_out.
//
// ISA fragment layouts (32-bit, wave32):
//   A 16x4 : lane L holds A[m=L&15][k = 2*(L>>4) + vgpr]        (2 VGPRs)
//   B 4x16 : lane L holds B[k = 2*(L>>4) + vgpr][n=L&15]        (2 VGPRs)
//   C 16x16: VGPR r, lanes 0-15 -> (M=r, N=lane); lanes 16-31 -> (M=8+r, N=lane-16)
//   diag(q): q<8 -> lane q, c[q];  q>=8 -> lane q+16, c[q-8]
// ---------------------------------------------------------------------------
__global__ __launch_bounds__(32)
void lgcn_score(const float* __restrict__ acc,
                const int* __restrict__ users,
                const int* __restrict__ items,
                const int* __restrict__ negs,
                float* __restrict__ out,
                int num_users, int batch) {
    __shared__ float s_pos[16];
    __shared__ float s_negexp[16];
    __shared__ float s_loss[16];

    const int lane = threadIdx.x;        // 0..31
    const int n    = lane & 15;          // matrix row/col handled by this lane
    const int kp   = lane >> 4;          // K-parity
    const int g    = blockIdx.x;         // group of 16 batch entries
    const int b    = g * 16 + n;

    // ---- A fragments: this group's user rows (kept in registers, reused 9x)
    const int urow = users[b];
    const float* up = acc + (size_t)urow * DIM + kp * 2;
    v2f afrag[16];
#pragma unroll
    for (int c = 0; c < 16; ++c)
        afrag[c] = *(const v2f*)(up + 4 * c);

    if (lane < 16) s_negexp[lane] = 0.0f;
    __syncthreads();

    const float scale = 1.0f / 16.0f;    // (1/(HOP+1))^2 = 1/16

    for (int t = 0; t < 9; ++t) {
        int vrow;
        if (t == 0) vrow = num_users + items[b];
        else        vrow = num_users + negs[(t - 1) * batch + b];

        const float* vp = acc + (size_t)vrow * DIM + kp * 2;
        v2f bfrag[16];
#pragma unroll
        for (int c = 0; c < 16; ++c)
            bfrag[c] = *(const v2f*)(vp + 4 * c);

        v8f cfrag = {};
#pragma unroll
        for (int c = 0; c < 16; ++c)
            cfrag = __builtin_amdgcn_wmma_f32_16x16x4_f32(
                /*neg_a=*/false, afrag[c],
                /*neg_b=*/false, bfrag[c],
                /*c_mod=*/(short)0, cfrag,
                /*reuse_a=*/false, /*reuse_b=*/false);

        // ---- extract diagonal element held by this lane (if any)
        const int  idx   = (lane < 8) ? lane : (lane - 24);
        const bool holds = (lane < 8) || (lane >= 24);
        float d = cfrag[0];
#pragma unroll
        for (int i = 1; i < 8; ++i)
            d = (idx == i) ? cfrag[i] : d;

        const int slot = (lane < 8) ? lane : (lane - 16);
        const float s  = d * scale;
        if (holds) {
            if (t == 0) s_pos[slot] = s;
            else        s_negexp[slot] += __expf(s);
        }
        __syncthreads();
    }

    // ---- per-entry loss: log(exp(pos) + sum_neg exp) - pos
    if (lane < 16) {
        float pos = s_pos[lane];
        s_loss[lane] = logf(__expf(pos) + s_negexp[lane]) - pos;
    }
    __syncthreads();

    if (lane == 0) {
        float sum = 0.0f;
#pragma unroll
        for (int i = 0; i < 16; ++i) sum += s_loss[i];
        atomicAdd(out, sum / (float)batch);
    }
}

// ---------------------------------------------------------------------------
// Host-side launch sequence (graph-capture safe: only kernels + memsetAsync)
// ---------------------------------------------------------------------------
extern "C" void kernel_launch(void* const* d_in, const int* in_sizes, int n_in,
                              void* d_out, int out_size, void* d_ws, size_t ws_size,
                              hipStream_t stream) {
    const float* user_emb = (const float*)d_in[0];
    const float* item_emb = (const float*)d_in[1];
    const float* A_vals   = (const float*)d_in[2];
    const int*   A_rows   = (const int*)d_in[3];
    const int*   A_cols   = (const int*)d_in[4];
    const int*   users    = (const int*)d_in[5];
    const int*   items    = (const int*)d_in[6];
    const int*   negs     = (const int*)d_in[7];

    const int num_users = in_sizes[0] / DIM;
    const int num_items = in_sizes[1] / DIM;
    const int nnz       = in_sizes[2];
    const int batch     = in_sizes[5];
    const int N         = num_users + num_items;
    const size_t Nd     = (size_t)N * DIM;          // floats per feature buffer

    float* acc = (float*)d_ws;
    float* xa  = acc + Nd;
    float* xb  = xa + Nd;

    // zero the scalar output (atomicAdd accumulates into it)
    hipMemsetAsync(d_out, 0, (size_t)out_size * sizeof(float), stream);

    // x0 = concat(user_emb, item_emb); acc = x0
    {
        const int total4 = (int)(Nd / 4);
        const int user4  = num_users * DIM / 4;
        const int blocks = (total4 + 255) / 256;
        lgcn_init_concat<<<blocks, 256, 0, stream>>>(
            (const float4*)user_emb, (const float4*)item_emb,
            (float4*)xa, (float4*)acc, user4, total4);
    }

    // 3 hops of x_{k+1} = A @ x_k ; acc += x_{k+1}
    for (int hop = 0; hop < HOPS; ++hop) {
        hipMemsetAsync(xb, 0, Nd * sizeof(float), stream);

        const int waves_per_block = 256 / 32;
        const int blocks = (nnz + waves_per_block - 1) / waves_per_block;
        lgcn_spmm<<<blocks, 256, 0, stream>>>(xa, xb, A_vals, A_rows, A_cols, nnz);

        const int total4 = (int)(Nd / 4);
        lgcn_accum<<<(total4 + 255) / 256, 256, 0, stream>>>(
            (float4*)acc, (const float4*)xb, total4);

        float* t = xa; xa = xb; xb = t;   // deterministic pointer swap
    }

    // WMMA scoring: one wave per 16 batch entries
    lgcn_score<<<batch / 16, 32, 0, stream>>>(
        acc, users, items, negs, (float*)d_out, num_users, batch);
}